// DiscriminativeLoss_66632122630437
// MI455X (gfx1250) — compile-verified
//
#include <hip/hip_runtime.h>
#include <hip/hip_bf16.h>

// Problem constants (match reference)
#define Bn   4
#define NFt  8
#define Hh   512
#define Ww   512
#define Cc   8
#define Npix (Hh * Ww)

#define DELTA_V 0.5f
#define DELTA_D 1.5f
#define ALPHA   1.0f
#define BETA    1.0f
#define GAMMA   0.001f

typedef __attribute__((ext_vector_type(2))) float v2f;
typedef __attribute__((ext_vector_type(8))) float v8f;

// Workspace layout (floats):
//   per-sample stats, stride 72: sums[8][8] (f-major), counts[8] at offset 64
//   then S[B] (per-sample variance distance sums) at offset B*72
#define STATS_STRIDE 72
#define S_OFF        (Bn * STATS_STRIDE)

// ---------------------------------------------------------------------------
// Pass 1: per-cluster sums + counts via V_WMMA_F32_16X16X4_F32.
// A (16x4): rows 0..7 = pred features for 4 pixels, row 8 = 1.0 (-> counts),
//           rows 9..15 = 0.  B (4x16): one-hot(label) per pixel row.
// D (16x16) accumulates sums[f][c] (M=0..7) and counts[c] (M=8).
// Branch-free inner loop: lanes with m>=8 load the clamped row (m&7), whose
// addresses coincide with lanes 0..7 (coalesced, no extra traffic), then
// v_cndmask selects the constant 1/0 rows. EXEC stays all-ones at the WMMA.
// ---------------------------------------------------------------------------
__global__ __launch_bounds__(128)
void dl_stats_kernel(const float* __restrict__ pred,
                     const int* __restrict__ tgt,
                     float* __restrict__ ws)
{
    const int b    = blockIdx.y;
    const int lid  = threadIdx.x & 31;
    const int wave = blockIdx.x * (blockDim.x >> 5) + (threadIdx.x >> 5);
    const int m    = lid & 15;   // A: row M; B: col N
    const int g    = lid >> 4;   // K-pair group: lanes 0-15 -> K=0,1; 16-31 -> K=2,3

    const float* pb = pred + (size_t)b * NFt * Npix;
    const int*   tb = tgt  + (size_t)b * Npix;

    const int pixPerWave = 1024;           // 256 waves/sample * 1024 = Npix
    int       pix    = wave * pixPerWave;
    const int pixEnd = pix + pixPerWave;

    const float* rowPtr = pb + (size_t)(m & 7) * Npix;  // clamped row: duplicate addrs coalesce
    const bool  isFeat = (m < NFt);
    const float cval   = (m == NFt) ? 1.0f : 0.0f;      // counts row / zero rows

    v8f acc = {};
    for (; pix < pixEnd; pix += 4) {
        const int p0 = pix + 2 * g;        // this lane's K-pair pixels

        const int2   lp = *(const int2*)  (tb     + p0);  // labels (b64)
        const float2 av = *(const float2*)(rowPtr + p0);  // pred pair (b64)

        v2f A  = { isFeat ? av.x : cval,
                   isFeat ? av.y : cval };
        v2f Bv = { (lp.x == m) ? 1.0f : 0.0f,
                   (lp.y == m) ? 1.0f : 0.0f };

        // D = A x B + D  (8 args: neg_a, A, neg_b, B, c_mod, C, reuse_a, reuse_b)
        acc = __builtin_amdgcn_wmma_f32_16x16x4_f32(
                  false, A, false, Bv, (short)0, acc, false, false);
    }

    // D[M][N]: lane L, vgpr r -> M = r + 8*(L>>4), N = L&15.
    float* st = ws + b * STATS_STRIDE;
    if (lid < Cc) {                                 // M=0..7 (sums), N=lid
        #pragma unroll
        for (int r = 0; r < NFt; ++r)
            atomicAdd(&st[r * Cc + lid], acc[r]);
    } else if (lid >= 16 && lid < 16 + Cc) {        // M=8 (counts), N=lid-16
        atomicAdd(&st[NFt * Cc + (lid - 16)], acc[0]);
    }
}

// ---------------------------------------------------------------------------
// Pass 2: S_b = sum_n clip(||mu_{g(n)} - p_n|| - delta_v, 0, 1e5)^2
// Means cached in LDS; 2 pixels per thread with b64 loads (fully coalesced:
// 8 B/lane contiguous); block tree-reduce, one global f32 atomic per block.
// ---------------------------------------------------------------------------
__global__ __launch_bounds__(256)
void dl_var_kernel(const float* __restrict__ pred,
                   const int* __restrict__ tgt,
                   float* __restrict__ ws)
{
    __shared__ float mu[NFt][Cc];
    __shared__ float red[256];

    const int b = blockIdx.y;
    const int t = threadIdx.x;

    const float* st = ws + b * STATS_STRIDE;
    if (t < NFt * Cc) {
        const int f = t >> 3, c = t & 7;
        mu[f][c] = st[f * Cc + c] / st[NFt * Cc + c];
    }
    __syncthreads();

    const float* pb = pred + (size_t)b * NFt * Npix;
    const int*   tb = tgt  + (size_t)b * Npix;

    float local = 0.0f;
    const int pixPerBlock = Npix / gridDim.x;       // 2048 with 128 blocks
    const int base = blockIdx.x * pixPerBlock;
    for (int k = 0; k < pixPerBlock; k += 2 * 256) {
        const int n = base + k + 2 * t;             // this thread's pixel pair
        const int2 lab = *(const int2*)(tb + n);
        float dsq0 = 0.0f, dsq1 = 0.0f;
        #pragma unroll
        for (int f = 0; f < NFt; ++f) {
            const float2 p = *(const float2*)(pb + (size_t)f * Npix + n);
            const float d0 = mu[f][lab.x] - p.x;
            const float d1 = mu[f][lab.y] - p.y;
            dsq0 += d0 * d0;
            dsq1 += d1 * d1;
        }
        const float n0 = sqrtf(dsq0);               // sqrt(0)=0: matches safe norm
        const float n1 = sqrtf(dsq1);
        const float v0 = fminf(fmaxf(n0 - DELTA_V, 0.0f), 100000.0f);
        const float v1 = fminf(fmaxf(n1 - DELTA_V, 0.0f), 100000.0f);
        local += v0 * v0 + v1 * v1;
    }

    red[t] = local;
    __syncthreads();
    for (int s = 128; s > 0; s >>= 1) {
        if (t < s) red[t] += red[t + s];
        __syncthreads();
    }
    if (t == 0) atomicAdd(&ws[S_OFF + b], red[0]);
}

// ---------------------------------------------------------------------------
// Pass 3: finalize, parallelized over 64 threads (one (i,j) cluster pair
// each for l_dist; row i==0 threads also handle l_reg / Sum(1/counts) / l_var)
// to avoid a serial single-thread tail.
// ---------------------------------------------------------------------------
__global__ __launch_bounds__(64)
void dl_final_kernel(const float* __restrict__ ws,
                     float* __restrict__ out)
{
    __shared__ float red[64];
    const int t = threadIdx.x;
    const int i = t >> 3;      // cluster i
    const int j = t & 7;       // cluster j

    float total = 0.0f;
    for (int b = 0; b < Bn; ++b) {
        const float* st = ws + b * STATS_STRIDE;
        const float cnti = st[NFt * Cc + i];
        const float cntj = st[NFt * Cc + j];

        float contrib = 0.0f;

        // l_dist contribution for pair (i, j), i != j; margin = 2*DELTA_D
        float sq = 0.0f;
        float sqj = 0.0f;   // ||mu_j||^2 (reused by i==0 threads for l_reg)
        #pragma unroll
        for (int f = 0; f < NFt; ++f) {
            const float mi = st[f * Cc + i] / cnti;
            const float mj = st[f * Cc + j] / cntj;
            const float d = mi - mj;
            sq  += d * d;
            sqj += mj * mj;
        }
        if (i != j) {
            const float nrm = (sq > 0.0f) ? sqrtf(sq) : 0.0f;
            const float v = fminf(fmaxf(2.0f * DELTA_D - nrm, 0.0f), 100000.0f);
            contrib += BETA * v * v / (float)(Cc * (Cc - 1));
        }
        if (i == 0) {
            // l_reg term for cluster j
            contrib += GAMMA * ((sqj > 0.0f) ? sqrtf(sqj) : 0.0f) / (float)Cc;
            // l_var (reference quirk): S_b * sum_c(1/counts_c) / C, distributed
            contrib += ALPHA * ws[S_OFF + b] * (1.0f / cntj) / (float)Cc;
        }

        red[t] = contrib;
        __syncthreads();
        for (int s = 32; s > 0; s >>= 1) {
            if (t < s) red[t] += red[t + s];
            __syncthreads();
        }
        if (t == 0) total += red[0];
        __syncthreads();
    }
    if (t == 0) out[0] = total / (float)Bn;
}

// ---------------------------------------------------------------------------
extern "C" void kernel_launch(void* const* d_in, const int* in_sizes, int n_in,
                              void* d_out, int out_size, void* d_ws, size_t ws_size,
                              hipStream_t stream)
{
    (void)in_sizes; (void)n_in; (void)out_size; (void)ws_size;
    const float* pred = (const float*)d_in[0];   // (B, NF, H, W) f32
    const int*   tgt  = (const int*)d_in[1];     // (B, H, W) i32
    float* out = (float*)d_out;
    float* ws  = (float*)d_ws;

    // Zero accumulators (graph-capturable memset node).
    hipMemsetAsync(d_ws, 0, (size_t)(S_OFF + Bn) * sizeof(float), stream);

    // Pass 1: 64 blocks x 4 waves = 256 waves per sample, 1024 px each.
    dim3 g1(64, Bn);
    dl_stats_kernel<<<g1, 128, 0, stream>>>(pred, tgt, ws);

    // Pass 2: 128 blocks per sample, 2048 px per block.
    dim3 g2(128, Bn);
    dl_var_kernel<<<g2, 256, 0, stream>>>(pred, tgt, ws);

    // Pass 3: finalize (64 threads, parallel pairs + reduce).
    dl_final_kernel<<<1, 64, 0, stream>>>(ws, out);
}